// SoftSkeletonizer3D_42623255446321
// MI455X (gfx1250) — compile-verified
//
#include <hip/hip_runtime.h>
#include <hip/hip_bf16.h>

// SoftSkeletonizer3D: 12 passes (2 iters x 6 directions) of a 26-neighbor
// boolean-polynomial thinning stencil on a 160^3 fp32 volume.
// CDNA5 paths: async global->LDS tile staging (global_load_async_to_lds_b32 /
// s_wait_asynccnt, ASYNCcnt-tracked) + packed-fp32 (v_pk_*) polynomial math,
// two x-adjacent voxels per thread. The polynomial is hand-factored into a
// balanced product tree with explicit CSE of subproducts shared across terms
// (the compiler would not reassociate the left-associated chains itself).

#define VOL_N 160
#define TILE  16
#define PT    18              // TILE + 2 (padded)
#define PT2   (PT * PT)       // 324
#define PVOL  (PT * PT * PT)  // 5832
#define NSTEP 12              // MAX_ITER(2) * 6 directions

typedef float v2f __attribute__((ext_vector_type(2)));

// Per-direction orderings of the 26 neighbors of the 3x3x3 cube (padded-cube
// coords [d,h,w] in {0,1,2}; center [1,1,1] excluded). Verbatim from reference.
__device__ constexpr signed char IND3[6][26][3] = {
 {{2,0,0},{2,0,1},{2,0,2},{1,0,2},{0,0,2},{0,0,1},{0,0,0},{1,0,0},{1,0,1},
  {2,1,0},{2,1,1},{2,1,2},{1,1,2},{0,1,2},{0,1,1},{0,1,0},{1,1,0},
  {2,2,0},{2,2,1},{2,2,2},{1,2,2},{0,2,2},{0,2,1},{0,2,0},{1,2,0},{1,2,1}},
 {{2,0,2},{2,1,2},{2,2,2},{1,2,2},{0,2,2},{0,1,2},{0,0,2},{1,0,2},{1,1,2},
  {2,0,1},{2,1,1},{2,2,1},{1,2,1},{0,2,1},{0,1,1},{0,0,1},{1,0,1},
  {2,0,0},{2,1,0},{2,2,0},{1,2,0},{0,2,0},{0,1,0},{0,0,0},{1,0,0},{1,1,0}},
 {{0,0,0},{0,0,1},{0,0,2},{0,1,2},{0,2,2},{0,2,1},{0,2,0},{0,1,0},{0,1,1},
  {1,0,0},{1,0,1},{1,0,2},{1,1,2},{1,2,2},{1,2,1},{1,2,0},{1,1,0},
  {2,0,0},{2,0,1},{2,0,2},{2,1,2},{2,2,2},{2,2,1},{2,2,0},{2,1,0},{2,1,1}},
 {{0,2,0},{0,2,1},{0,2,2},{1,2,2},{2,2,2},{2,2,1},{2,2,0},{1,2,0},{1,2,1},
  {0,1,0},{0,1,1},{0,1,2},{1,1,2},{2,1,2},{2,1,1},{2,1,0},{1,1,0},
  {0,0,0},{0,0,1},{0,0,2},{1,0,2},{2,0,2},{2,0,1},{2,0,0},{1,0,0},{1,0,1}},
 {{2,2,0},{2,1,0},{2,0,0},{1,0,0},{0,0,0},{0,1,0},{0,2,0},{1,2,0},{1,1,0},
  {2,2,1},{2,1,1},{2,0,1},{1,0,1},{0,0,1},{0,1,1},{0,2,1},{1,2,1},
  {2,2,2},{2,1,2},{2,0,2},{1,0,2},{0,0,2},{0,1,2},{0,2,2},{1,2,2},{1,1,2}},
 {{2,2,0},{2,2,1},{2,2,2},{2,1,2},{2,0,2},{2,0,1},{2,0,0},{2,1,0},{2,1,1},
  {1,2,0},{1,2,1},{1,2,2},{1,1,2},{1,0,2},{1,0,1},{1,0,0},{1,1,0},
  {0,2,0},{0,2,1},{0,2,2},{0,1,2},{0,0,2},{0,0,1},{0,0,0},{0,1,0},{0,1,1}},
};

// Packed (2-wide fp32) keep-mask polynomial, hand-factored:
//  - pair/quad product layers shared across M1..M6
//  - balanced trees (depth ~4 instead of ~15 for the 16-term chain)
__device__ __forceinline__ v2f keep_mask2(const v2f* __restrict__ n) {
  v2f q[26];
#pragma unroll
  for (int i = 0; i < 26; ++i) q[i] = 1.0f - n[i];

  // ---- shared pair products -------------------------------------------
  const v2f a01 = q[0]*q[1],  a23 = q[2]*q[3],  a34 = q[3]*q[4];
  const v2f a45 = q[4]*q[5],  a67 = q[6]*q[7];

  const v2f b910  = q[9]*q[10],  b1112 = q[11]*q[12];
  const v2f b1314 = q[13]*q[14], b1516 = q[15]*q[16];
  const v2f b1718 = q[17]*q[18], b1920 = q[19]*q[20];
  const v2f b2122 = q[21]*q[22], b2324 = q[23]*q[24];

  const v2f d1011 = q[10]*q[11], d1213 = q[12]*q[13];
  const v2f d1415 = q[14]*q[15], d1617 = q[16]*q[17];
  const v2f d1819 = q[18]*q[19], d2021 = q[20]*q[21];
  const v2f d2223 = q[22]*q[23];
  const v2f f121314 = d1213*q[14];

  const v2f c13 = q[1]*q[3], c57 = q[5]*q[7];
  const v2f c04 = q[0]*q[4], c46 = q[4]*q[6], c02 = q[0]*q[2];

  // ---- shared triple/quad products ------------------------------------
  const v2f t567 = q[5]*a67;     // q5 q6 q7
  const v2f t345 = q[3]*a45;     // q3 q4 q5
  const v2f t123 = q[1]*a23;     // q1 q2 q3
  const v2f t017 = a01*q[7];     // q0 q1 q7

  const v2f q0123 = a01*a23;
  const v2f q4567 = a45*a67;
  const v2f q07 = q0123*q4567;   // q0..q7
  const v2f q08 = q07*q[8];      // q0..q8
  const v2f s8  = q[8]*n[25];
  const v2f c9  = q08*q[25];

  const v2f e912  = b910*b1112;    // q9..q12
  const v2f e1316 = b1314*b1516;   // q13..q16
  const v2f e1720 = b1718*b1920;   // q17..q20
  const v2f e2124 = b2122*b2324;   // q21..q24
  const v2f p9_24 = (e912*e1316)*(e1720*e2124);

  // ---- shared n-pair products ------------------------------------------
  const v2f w1012 = n[10]*n[12], w1016 = n[10]*n[16];
  const v2f w1614 = n[16]*n[14], w1412 = n[14]*n[12];
  const v2f g1820 = n[18]*n[20], g1824 = n[18]*n[24];
  const v2f g2422 = n[24]*n[22], g2220 = n[22]*n[20];

  // ---- M1 ---------------------------------------------------------------
  const v2f M1 = (q07*s8) * (1.0f - p9_24);

  // ---- M2 ---------------------------------------------------------------
  const v2f m2a = 1.0f - (a34*t567)*n[10];     // q3q4q5q6q7 n10
  const v2f m2b = 1.0f - (t123*a45)*n[16];     // q1q2q3q4q5 n16
  const v2f m2c = 1.0f - (q0123*q[7])*n[14];   // q0q1q2q3q7 n14
  const v2f m2d = 1.0f - (a01*t567)*n[12];     // q0q1q5q6q7 n12
  const v2f M2 = s8 * (1.0f - (m2a*m2b)*(m2c*m2d));

  // ---- M3 ---------------------------------------------------------------
  const v2f m3a = 1.0f - t567*w1012;
  const v2f m3b = 1.0f - t345*w1016;
  const v2f m3c = 1.0f - t123*w1614;
  const v2f m3d = 1.0f - t017*w1412;
  const v2f M3 = s8 * (1.0f - (m3a*m3b)*(m3c*m3d));

  // ---- M4 ---------------------------------------------------------------
  const v2f m4a = 1.0f - (c04*q[6])*(n[11]*n[2]);
  const v2f m4b = 1.0f - (c46*q[2])*(n[0]*n[9]);
  const v2f m4c = 1.0f - (c04*q[2])*(n[6]*n[15]);
  const v2f m4d = 1.0f - (c02*q[6])*(n[4]*n[13]);
  const v2f M4 = ((c13*c57)*s8) * (1.0f - (m4a*m4b)*(m4c*m4d));

  // ---- M5 ---------------------------------------------------------------
  const v2f Aout = (b1314*q[15])*(b2122*q[23]);       // q13q14q15 q21q22q23
  const v2f Ain  = (e912*d1617)*(b1920*q[24]);        // q9..q12 q16q17 q19q20 q24
  const v2f A    = 1.0f - (Aout*n[18])*(1.0f - Ain);

  const v2f Bout = (b1112*q[13])*(b1920*q[21]);       // q11q12q13 q19q20q21
  const v2f Bin  = ((b910*d1415)*(d1617*q[18]))*d2223;// q9q10 q14..q18 q22q23
  const v2f B    = 1.0f - (Bout*n[24])*(1.0f - Bin);

  const v2f Cout = (b910*q[11])*(b1718*q[19]);        // q9q10q11 q17q18q19
  const v2f Cin  = ((d1213*d1415)*q[16])*(d2021*b2324); // q12..q16 q20q21 q23q24
  const v2f C    = 1.0f - (Cout*n[22])*(1.0f - Cin);

  const v2f Dout = (q[9]*b1516)*(q[17]*b2324);        // q9 q15q16 q17 q23q24
  const v2f Din  = (d1011*f121314)*(d1819*b2122);     // q10..q14 q18q19 q21q22
  const v2f D    = 1.0f - (Dout*n[20])*(1.0f - Din);

  const v2f M5 = c9 * (1.0f - (A*B)*(C*D));

  // ---- M6 ---------------------------------------------------------------
  const v2f A2 = 1.0f - ((d1415*q[16])*(d2223*q[24]))*g1820;   // q14q15q16 q22q23q24
  const v2f B2 = 1.0f - (f121314*(d2021*q[22]))*g1824;         // q12q13q14 q20q21q22
  const v2f C2 = 1.0f - ((d1011*q[12])*(d1819*q[20]))*g2422;   // q10q11q12 q18q19q20
  const v2f D2 = 1.0f - ((b910*q[16])*(b1718*q[24]))*g2220;    // q9q10q16 q17q18q24
  const v2f M6 = c9 * (1.0f - (A2*B2)*(C2*D2));

  return ((1.0f - M1)*(1.0f - M2))
       * ((1.0f - M3)*(1.0f - M4))
       * ((1.0f - M5)*(1.0f - M6));
}

__device__ __forceinline__ void async_load_lds_b32(unsigned lds_byte,
                                                   const float* g) {
  const unsigned long long gaddr = (unsigned long long)(const void*)g;
  asm volatile("global_load_async_to_lds_b32 %0, %1, off"
               :
               : "v"(lds_byte), "v"(gaddr)
               : "memory");
}

// One thinning pass for direction O. 16^3 output tile per 256-thread block;
// padded 18^3 tile staged into LDS via CDNA5 async global->LDS copies.
// Each thread evaluates 2 x-adjacent voxels with packed-fp32 math.
template<int O>
__global__ __launch_bounds__(256)
void skel_step(const float* __restrict__ src, float* __restrict__ dst) {
  __shared__ float t[PVOL];  // single LDS alloc -> byte offset of t[0] is 0

  const int tid = threadIdx.x;
  const int x0 = blockIdx.x * TILE;
  const int y0 = blockIdx.y * TILE;
  const int z0 = blockIdx.z * TILE;

  // Uniform (scalar-branch) interior test: halo never leaves the volume.
  const bool interior =
      (blockIdx.x != 0) & (blockIdx.x != gridDim.x - 1) &
      (blockIdx.y != 0) & (blockIdx.y != gridDim.y - 1) &
      (blockIdx.z != 0) & (blockIdx.z != gridDim.z - 1);

  if (interior) {
    // Fast path: no zero-fill, no bounds checks; pure async staging.
    for (int i = tid; i < PVOL; i += 256) {
      const int px = i % PT;
      const int py = (i / PT) % PT;
      const int pz = i / PT2;
      const size_t gidx = ((size_t)(z0 + pz - 1) * VOL_N + (y0 + py - 1)) * VOL_N
                        + (x0 + px - 1);
      async_load_lds_b32((unsigned)(i * 4), src + gidx);
    }
  } else {
    // Boundary path: pre-zero the padded tile (zero-pad halo), then
    // async-copy only in-volume elements (EXEC-masked).
    for (int i = tid; i < PVOL; i += 256) t[i] = 0.0f;
    __syncthreads();
    for (int i = tid; i < PVOL; i += 256) {
      const int px = i % PT;
      const int py = (i / PT) % PT;
      const int pz = i / PT2;
      const int gx = x0 + px - 1;
      const int gy = y0 + py - 1;
      const int gz = z0 + pz - 1;
      if ((unsigned)gx < (unsigned)VOL_N &&
          (unsigned)gy < (unsigned)VOL_N &&
          (unsigned)gz < (unsigned)VOL_N) {
        const size_t gidx = ((size_t)gz * VOL_N + gy) * VOL_N + gx;
        async_load_lds_b32((unsigned)(i * 4), src + gidx);
      }
    }
  }
  // Wait for this wave's async copies to land in LDS, then sync all waves.
  asm volatile("s_wait_asynccnt 0" ::: "memory");
  __syncthreads();

  // Compute: 8 x-pairs x 16 y x 2 z-halves = 256 threads;
  // each thread marches 8 z-planes, 2 voxels (packed fp32) per plane.
  const int px2 = tid & 7;          // x-pair index (x = 2*px2, 2*px2+1)
  const int py  = (tid >> 3) & 15;  // y within tile
  const int zh  = tid >> 7;         // z half (0 or 1)
  const int lx  = 2 * px2 + 1;      // LDS x of first voxel of the pair
  const int ly  = py + 1;

  for (int i8 = 0; i8 < 8; ++i8) {
    const int tz = zh * 8 + i8;
    const int base = (tz + 1) * PT2 + ly * PT + lx;
    v2f nb[26];
#pragma unroll
    for (int k = 0; k < 26; ++k) {
      // Compile-time neighbor offsets; each pair is two consecutive LDS words
      // (fused into paired/64-bit DS loads by the compiler).
      const int off = ((int)IND3[O][k][0] - 1) * PT2
                    + ((int)IND3[O][k][1] - 1) * PT
                    + ((int)IND3[O][k][2] - 1);
      v2f v = { t[base + off], t[base + off + 1] };
      nb[k] = v;
    }
    v2f center = { t[base], t[base + 1] };
    const v2f keep = keep_mask2(nb);
    const v2f r = center * keep;

    // 8-byte aligned packed store (x even, row stride 640B).
    float* dptr = dst + ((size_t)(z0 + tz) * VOL_N + (y0 + py)) * VOL_N
                + (x0 + 2 * px2);
    *(v2f*)dptr = r;
  }
}

extern "C" void kernel_launch(void* const* d_in, const int* in_sizes, int n_in,
                              void* d_out, int out_size, void* d_ws, size_t ws_size,
                              hipStream_t stream) {
  const float* in = (const float*)d_in[0];
  float* out = (float*)d_out;
  float* ws  = (float*)d_ws;   // one 160^3 fp32 volume (16.4 MB) of scratch

  const dim3 grid(VOL_N / TILE, VOL_N / TILE, VOL_N / TILE);
  const dim3 block(256);

  const float* src = in;
  for (int s = 0; s < NSTEP; ++s) {
    float* dst = (s & 1) ? out : ws;   // step 11 (last) lands in d_out
    switch (s % 6) {
      case 0: skel_step<0><<<grid, block, 0, stream>>>(src, dst); break;
      case 1: skel_step<1><<<grid, block, 0, stream>>>(src, dst); break;
      case 2: skel_step<2><<<grid, block, 0, stream>>>(src, dst); break;
      case 3: skel_step<3><<<grid, block, 0, stream>>>(src, dst); break;
      case 4: skel_step<4><<<grid, block, 0, stream>>>(src, dst); break;
      case 5: skel_step<5><<<grid, block, 0, stream>>>(src, dst); break;
    }
    src = dst;
  }
}